// tensor_DA_layer_75316546503011
// MI455X (gfx1250) — compile-verified
//
#include <hip/hip_runtime.h>
#include <stdint.h>

// Problem constants (fixed by the reference).
#define N_UNIT 1024
#define N_LOAD 2048
#define TPB 256            // threads per block in the streaming kernel
#define ROWS_PER_BLOCK 32  // output rows per block
#define PREV_OFF 256       // float offset of prev[] inside workspace

typedef __attribute__((ext_vector_type(4))) float v4f;

// Pointer types matching the async-to-LDS builtin prototype:
//   param0: int4 in global (AS1 / __device__), param1: int4 in LDS (AS3 / __shared__)
typedef int v4i __attribute__((vector_size(16)));
typedef __attribute__((address_space(1))) v4i* gv4i_p;
typedef __attribute__((address_space(3))) v4i* lv4i_p;
typedef __attribute__((address_space(3))) void lvoid;

// ---- CDNA5 async global->LDS copy (ASYNCcnt-tracked) -----------------------
__device__ __forceinline__ void async_copy_b128_to_lds(void* lds_dst, const void* gsrc) {
#if __has_builtin(__builtin_amdgcn_global_load_async_to_lds_b128)
  __builtin_amdgcn_global_load_async_to_lds_b128(
      (gv4i_p)(uintptr_t)gsrc,   // global source (AS1)
      (lv4i_p)lds_dst,           // LDS destination (AS3, addrspacecast)
      /*offset=*/0, /*cpol=*/0);
#else
  // CDNA5 ISA: GLOBAL_LOAD_ASYNC_TO_LDS_B128, GV mode.
  unsigned int lds_off = (unsigned int)(uintptr_t)(lvoid*)lds_dst;
  asm volatile("global_load_async_to_lds_b128 %0, %1, off"
               :: "v"(lds_off), "v"(gsrc)
               : "memory");
#endif
}

__device__ __forceinline__ void wait_asynccnt0() {
#if __has_builtin(__builtin_amdgcn_s_wait_asynccnt)
  __builtin_amdgcn_s_wait_asynccnt(0);
#else
  asm volatile("s_wait_asynccnt 0" ::: "memory");
#endif
}

// ---- Phase 1: rank by Cost, exclusive cumsum of sorted Pmax, sum(Pd) -------
// One block of 1024 threads. ws[0] = sum(Pd); ws[PREV_OFF + i] = prev for unit i.
__global__ __launch_bounds__(N_UNIT) void ed_setup_kernel(
    const float* __restrict__ Cost, const float* __restrict__ Pmax,
    const float* __restrict__ Pd, float* __restrict__ ws) {
  __shared__ float sCost[N_UNIT];
  __shared__ float sScan[N_UNIT];
  __shared__ float sRed[N_UNIT];

  const int t = threadIdx.x;
  sCost[t] = Cost[t];
  const float pm = Pmax[t];
  sRed[t] = Pd[t] + Pd[t + N_UNIT];
  __syncthreads();

  // Stable ascending rank: count of strictly-smaller, ties broken by index.
  const float ci = sCost[t];
  int rank = 0;
  #pragma unroll 8
  for (int j = 0; j < N_UNIT; ++j) {
    const float cj = sCost[j];
    rank += (cj < ci || (cj == ci && j < t)) ? 1 : 0;
  }

  // Scatter Pmax into merit order (ranks are a permutation -> no collisions).
  sScan[rank] = pm;
  __syncthreads();

  // Hillis-Steele inclusive scan over sorted Pmax.
  for (int off = 1; off < N_UNIT; off <<= 1) {
    const float add = (t >= off) ? sScan[t - off] : 0.0f;
    __syncthreads();
    sScan[t] += add;
    __syncthreads();
  }

  // prev (in ORIGINAL unit order) = cum[rank] - Pmax_sorted[rank]  (== reference).
  ws[PREV_OFF + t] = sScan[rank] - pm;

  // Tree-reduce sum(Pd).
  for (int s = N_UNIT / 2; s > 0; s >>= 1) {
    __syncthreads();
    if (t < s) sRed[t] += sRed[t + s];
  }
  __syncthreads();
  if (t == 0) ws[0] = sRed[0];
}

// ---- Phase 2: stream 256 MB of clamped outputs -----------------------------
// Block = 256 threads covering all 1024 units (4 each), ROWS_PER_BLOCK rows.
// prev/Pmax staged via async global->LDS once per block, then held in regs.
__global__ __launch_bounds__(TPB) void ed_dispatch_kernel(
    const float* __restrict__ x, const float* __restrict__ Pmax,
    const float* __restrict__ wcap, const float* __restrict__ ws,
    float* __restrict__ out) {
  __shared__ __align__(16) float sPrev[N_UNIT];
  __shared__ __align__(16) float sPmax[N_UNIT];

  const int t = threadIdx.x;
  const int i0 = t * 4;

  // Async-stage per-unit data into LDS (2x b128 per thread = 8 KB total).
  async_copy_b128_to_lds(&sPrev[i0], ws + PREV_OFF + i0);
  async_copy_b128_to_lds(&sPmax[i0], Pmax + i0);

  // Uniform scalars (scalar loads) while the async copies are in flight.
  const float sumPd = ws[0];
  const float cap = wcap[0];

  wait_asynccnt0();
  __syncthreads();

  const v4f pv = *(const v4f*)(&sPrev[i0]);
  const v4f pm = *(const v4f*)(&sPmax[i0]);

  const size_t b0 = (size_t)blockIdx.x * ROWS_PER_BLOCK;
  float* orow = out + b0 * (size_t)N_UNIT + (size_t)i0;

  #pragma unroll 4
  for (int r = 0; r < ROWS_PER_BLOCK; ++r) {
    const float td = sumPd - cap * x[b0 + r];  // block-uniform -> SALU/SMEM
    v4f o;
    o.x = fminf(fmaxf(td - pv.x, 0.0f), pm.x);
    o.y = fminf(fmaxf(td - pv.y, 0.0f), pm.y);
    o.z = fminf(fmaxf(td - pv.z, 0.0f), pm.z);
    o.w = fminf(fmaxf(td - pv.w, 0.0f), pm.w);
    // Streaming output: written once, never re-read on device -> non-temporal.
    __builtin_nontemporal_store(o, (v4f*)orow);
    orow += N_UNIT;
  }
}

extern "C" void kernel_launch(void* const* d_in, const int* in_sizes, int n_in,
                              void* d_out, int out_size, void* d_ws, size_t ws_size,
                              hipStream_t stream) {
  (void)n_in; (void)out_size; (void)ws_size;
  const float* x    = (const float*)d_in[0];
  const float* Cost = (const float*)d_in[1];
  const float* Pmax = (const float*)d_in[2];
  const float* Pd   = (const float*)d_in[3];
  const float* wcap = (const float*)d_in[4];
  float* out = (float*)d_out;
  float* ws  = (float*)d_ws;  // needs (PREV_OFF + N_UNIT) * 4 = 5 KB

  ed_setup_kernel<<<1, N_UNIT, 0, stream>>>(Cost, Pmax, Pd, ws);

  const int B = in_sizes[0];  // 65536, divisible by ROWS_PER_BLOCK
  ed_dispatch_kernel<<<B / ROWS_PER_BLOCK, TPB, 0, stream>>>(x, Pmax, wcap, ws, out);
}